// SparseKNNGraphAttention_26757646254308
// MI455X (gfx1250) — compile-verified
//
#include <hip/hip_runtime.h>

typedef float v2f __attribute__((ext_vector_type(2)));
typedef float v8f __attribute__((ext_vector_type(8)));

#define DMODEL 256
#define ROWS_PER_BLOCK 128   // 8 waves * 16 rows
#define COL_SPLIT 4          // 4 col-quarters of 4 tiles each
#define BS_STRIDE 260        // 256 + 4 pad -> conflict-free ds_load_b64 B-fragments

// ---------------------------------------------------------------------------
// Meff[n][d] = (n==d) + sum_t Wout[n][t] * Wqkv[(512+t)][d]
// (identity folds the residual "x +" into the single GEMM)
// ---------------------------------------------------------------------------
__global__ __launch_bounds__(256) void build_meff(const float* __restrict__ Wqkv,
                                                  const float* __restrict__ Wout,
                                                  float* __restrict__ Meff) {
    const int n = blockIdx.x;     // output row (uniform per block -> scalar loads)
    const int d = threadIdx.x;    // output col (contiguous per thread)
    float acc = (n == d) ? 1.0f : 0.0f;
    const float* wrow = Wout + n * DMODEL;
    #pragma unroll 8
    for (int t = 0; t < DMODEL; ++t)
        acc = fmaf(wrow[t], Wqkv[(512 + t) * DMODEL + d], acc);
    Meff[n * DMODEL + d] = acc;
}

// c[n] = bout[n] + sum_t Wout[n][t] * bqkv[512+t]
__global__ __launch_bounds__(256) void build_bias(const float* __restrict__ Wout,
                                                  const float* __restrict__ bqkv,
                                                  const float* __restrict__ bout,
                                                  float* __restrict__ cvec) {
    const int n = threadIdx.x;
    float acc = bout[n];
    #pragma unroll 8
    for (int t = 0; t < DMODEL; ++t)
        acc = fmaf(Wout[n * DMODEL + t], bqkv[512 + t], acc);
    cvec[n] = acc;
}

// ---------------------------------------------------------------------------
// y[m][n] = sum_d x[m][d] * Meff[n][d] + c[n]
// WMMA F32 16x16x4: A = x strip (reg-resident, ISA 16x4 layout),
//                   B = Meff slice from LDS, C/D = f32 accumulators.
// ---------------------------------------------------------------------------
__global__ __launch_bounds__(256) void fused_gemm(const float* __restrict__ x,
                                                  const float* __restrict__ Meff,
                                                  const float* __restrict__ cvec,
                                                  float* __restrict__ y) {
    __shared__ float Bs[16 * BS_STRIDE];

    const int lane    = threadIdx.x & 31;
    const int wave    = threadIdx.x >> 5;
    const int half    = lane >> 4;    // ISA: lanes 16-31 carry K+2 / M+8
    const int l16     = lane & 15;
    const int rowBase = blockIdx.x * ROWS_PER_BLOCK;
    const int colBase = blockIdx.y * (16 / COL_SPLIT);   // in 16-col tiles

    // --- A strip: register-resident in WMMA 16x4-f32 layout ---------------
    // areg[kk] holds x[row][4*kk + 2*half + {0,1}]  (global_load_b64 each)
    const int mrow = rowBase + wave * 16 + l16;
    const float* arow = x + mrow * DMODEL + 2 * half;
    v2f areg[64];
    #pragma unroll
    for (int kk = 0; kk < 64; ++kk)
        areg[kk] = *reinterpret_cast<const v2f*>(arow + 4 * kk);

    // cooperative-staging coordinates: 256 threads move 16 rows x 256 floats
    const int sRow = threadIdx.x >> 4;   // which Meff row of the tile
    const int sQ   = threadIdx.x & 15;   // which 16-float chunk of that row

    for (int t = 0; t < 16 / COL_SPLIT; ++t) {
        const int ncol = colBase + t;    // 16-col tile index

        __syncthreads();                 // previous tile's B reads done
        {   // stage Meff rows [16*ncol, 16*ncol+16) into padded LDS
            const float* src = Meff + (ncol * 16 + sRow) * DMODEL + sQ * 16;
            float*       dst = Bs + sRow * BS_STRIDE + sQ * 16;
            #pragma unroll
            for (int j = 0; j < 4; ++j)
                reinterpret_cast<float4*>(dst)[j] =
                    reinterpret_cast<const float4*>(src)[j];
        }
        __syncthreads();

        // --- 64 x v_wmma_f32_16x16x4_f32, two accumulators for ILP --------
        v8f acc0 = {};
        v8f acc1 = {};
        const float* brow = Bs + l16 * BS_STRIDE + 2 * half;  // B[k][n=l16]
        #pragma unroll
        for (int kk = 0; kk < 64; kk += 2) {
            v2f b0 = *reinterpret_cast<const v2f*>(brow + 4 * kk);
            v2f b1 = *reinterpret_cast<const v2f*>(brow + 4 * (kk + 1));
            acc0 = __builtin_amdgcn_wmma_f32_16x16x4_f32(
                false, areg[kk],     false, b0, (short)0, acc0, false, false);
            acc1 = __builtin_amdgcn_wmma_f32_16x16x4_f32(
                false, areg[kk + 1], false, b1, (short)0, acc1, false, false);
        }
        v8f acc = acc0 + acc1;

        // --- epilogue: + c[col], coalesced stores (C/D ISA layout) --------
        const int col  = ncol * 16 + l16;
        const int row0 = rowBase + wave * 16 + 8 * half;
        const float cv = cvec[col];
        #pragma unroll
        for (int g = 0; g < 8; ++g)
            y[(row0 + g) * DMODEL + col] = acc[g] + cv;
    }
}

// ---------------------------------------------------------------------------
extern "C" void kernel_launch(void* const* d_in, const int* in_sizes, int n_in,
                              void* d_out, int out_size, void* d_ws, size_t ws_size,
                              hipStream_t stream) {
    const float* x    = (const float*)d_in[0];  // [4,4096,256]
    const float* Wqkv = (const float*)d_in[1];  // [768,256]
    const float* bqkv = (const float*)d_in[2];  // [768]
    const float* Wout = (const float*)d_in[3];  // [256,256]
    const float* bout = (const float*)d_in[4];  // [256]
    float* y = (float*)d_out;                   // [4,4096,256]

    float* Meff = (float*)d_ws;                 // 256*256 floats
    float* cvec = Meff + DMODEL * DMODEL;       // 256 floats

    build_meff<<<DMODEL, DMODEL, 0, stream>>>(Wqkv, Wout, Meff);
    build_bias<<<1, DMODEL, 0, stream>>>(Wout, bqkv, bout, cvec);

    const int totalRows = 4 * 4096;             // B*N
    dim3 grid(totalRows / ROWS_PER_BLOCK, COL_SPLIT);
    fused_gemm<<<grid, 256, 0, stream>>>(x, Meff, cvec, y);
}